// GCN_68985764708480
// MI455X (gfx1250) — compile-verified
//
#include <hip/hip_runtime.h>
#include <hip/hip_bf16.h>

typedef __attribute__((ext_vector_type(2))) float v2f;
typedef __attribute__((ext_vector_type(8))) float v8f;

#define IN_C 256
#define HIDC 16

// ---------------------------------------------------------------------------
// Degree / norm precompute (graph-only, done once, reused by all 3 layers)
// ---------------------------------------------------------------------------
__global__ void init_deg_kernel(float* __restrict__ deg, int n) {
    int i = blockIdx.x * blockDim.x + threadIdx.x;
    if (i < n) deg[i] = 1.0f;                 // self-loop
}

__global__ void count_deg_kernel(const int* __restrict__ dst, float* __restrict__ deg, int e) {
    int i = blockIdx.x * blockDim.x + threadIdx.x;
    if (i < e) atomicAdd(&deg[dst[i]], 1.0f);
}

__global__ void deg_to_dinv_kernel(float* __restrict__ deg, int n) {
    int i = blockIdx.x * blockDim.x + threadIdx.x;
    if (i < n) deg[i] = rsqrtf(deg[i]);       // deg >= 1 always (self-loops)
}

// ---------------------------------------------------------------------------
// Tall-skinny fp32 GEMM with V_WMMA_F32_16X16X4_F32.
// One wave32 per 16-row tile; Fout fixed at 16; K = 256 or 16.
// A tile staged in LDS (coalesced float4 loads, padded stride -> no bank
// conflicts on column reads, rows stay 16B aligned: (K+4)*4 % 16 == 0).
// ---------------------------------------------------------------------------
template <int K>
__global__ void __launch_bounds__(32) gemm16_wmma_kernel(const float* __restrict__ A,
                                                         const float* __restrict__ W,
                                                         float* __restrict__ C) {
    constexpr int LDK = K + 4;
    __shared__ float tile[16 * LDK];

    const int m0   = blockIdx.x * 16;
    const int lane = threadIdx.x;

    // Coalesced stage of the 16xK fp32 tile into LDS.
    const float4* Ag = (const float4*)(A + (size_t)m0 * K);
    #pragma unroll 4
    for (int i = lane; i < 16 * K / 4; i += 32) {
        int row = (i * 4) / K;
        int col = (i * 4) % K;
        *(float4*)&tile[row * LDK + col] = Ag[i];
    }
    __syncthreads();

    const int half = lane >> 4;   // 0: K={k,k+1}, 1: K={k+2,k+3}
    const int l    = lane & 15;

    v8f acc = {};
    #pragma unroll 4
    for (int k = 0; k < K; k += 4) {
        const int ka = k + half * 2;
        v2f a, b;
        a.x = tile[l * LDK + ka];         // A: M=l (row), K=ka
        a.y = tile[l * LDK + ka + 1];     //    K=ka+1
        b.x = W[ka * 16 + l];             // B: K=ka,   N=l
        b.y = W[(ka + 1) * 16 + l];       //    K=ka+1, N=l
        acc = __builtin_amdgcn_wmma_f32_16x16x4_f32(
            /*neg_a=*/false, a, /*neg_b=*/false, b,
            /*c_mod=*/(short)0, acc, /*reuse_a=*/false, /*reuse_b=*/false);
    }

    // D layout: VGPR r -> row (m0 + half*8 + r), col l.
    float* out = C + (size_t)(m0 + half * 8) * 16 + l;
    #pragma unroll
    for (int r = 0; r < 8; ++r) out[r * 16] = acc[r];
}

// ---------------------------------------------------------------------------
// Aggregation: out[i] = h[i]/deg[i] + bias  (self-loop + bias init), then
// edge scatter out[dst] += h[src] * dinv[src]*dinv[dst] via f32 atomics.
// ---------------------------------------------------------------------------
__global__ void init_agg_kernel(const float* __restrict__ h, const float* __restrict__ dinv,
                                const float* __restrict__ bias, float* __restrict__ agg, int n) {
    int i = blockIdx.x * blockDim.x + threadIdx.x;
    if (i >= n * 16) return;
    int node = i >> 4, c = i & 15;
    float di = dinv[node];
    agg[i] = h[i] * di * di + bias[c];
}

__global__ void scatter_edges_kernel(const int* __restrict__ src, const int* __restrict__ dst,
                                     const float* __restrict__ h, const float* __restrict__ dinv,
                                     float* __restrict__ agg, int e) {
    int t = blockIdx.x * blockDim.x + threadIdx.x;
    int edge = t >> 2, chunk = t & 3;       // 4 threads / edge, 4 channels each
    if (edge >= e) return;
    int s = src[edge], d = dst[edge];
    float w = dinv[s] * dinv[d];
    float4 v = *((const float4*)(h + (size_t)s * 16) + chunk);
    float* ad = agg + (size_t)d * 16 + (size_t)chunk * 4;
    atomicAdd(ad + 0, v.x * w);
    atomicAdd(ad + 1, v.y * w);
    atomicAdd(ad + 2, v.z * w);
    atomicAdd(ad + 3, v.w * w);
}

// ---------------------------------------------------------------------------
// BatchNorm: per-channel sum / sumsq (grid-stride, channel fixed per thread
// since stride is a multiple of 16), then fused normalize + ReLU.
// ---------------------------------------------------------------------------
__global__ void zero_stats_kernel(float* __restrict__ stats) {
    if (threadIdx.x < 32) stats[threadIdx.x] = 0.0f;
}

__global__ void bn_reduce_kernel(const float* __restrict__ x, float* __restrict__ stats, int total) {
    __shared__ float ssum[16], ssq[16];
    int t = threadIdx.x;
    if (t < 16) { ssum[t] = 0.0f; ssq[t] = 0.0f; }
    __syncthreads();
    int i = blockIdx.x * blockDim.x + t;
    int stride = blockDim.x * gridDim.x;    // multiple of 16 -> channel invariant
    int c = i & 15;
    float s = 0.0f, q = 0.0f;
    for (; i < total; i += stride) { float v = x[i]; s += v; q += v * v; }
    atomicAdd(&ssum[c], s);
    atomicAdd(&ssq[c], q);
    __syncthreads();
    if (t < 16) {
        atomicAdd(&stats[t], ssum[t]);
        atomicAdd(&stats[16 + t], ssq[t]);
    }
}

__global__ void bn_apply_relu_kernel(float* __restrict__ x, const float* __restrict__ stats,
                                     const float* __restrict__ gamma, const float* __restrict__ beta,
                                     int total, float invN) {
    int i = blockIdx.x * blockDim.x + threadIdx.x;
    if (i >= total) return;
    int c = i & 15;
    float mu  = stats[c] * invN;
    float var = stats[16 + c] * invN - mu * mu;   // biased variance (jnp.var)
    float sc  = gamma[c] * rsqrtf(var + 1e-5f);
    float v   = (x[i] - mu) * sc + beta[c];
    x[i] = v > 0.0f ? v : 0.0f;
}

// ---------------------------------------------------------------------------
extern "C" void kernel_launch(void* const* d_in, const int* in_sizes, int n_in,
                              void* d_out, int out_size, void* d_ws, size_t ws_size,
                              hipStream_t stream) {
    const float* x   = (const float*)d_in[0];
    const int*   ei  = (const int*)d_in[1];
    const float* W1  = (const float*)d_in[2];
    const float* b1  = (const float*)d_in[3];
    const float* g1  = (const float*)d_in[4];
    const float* be1 = (const float*)d_in[5];
    const float* W2  = (const float*)d_in[6];
    const float* b2  = (const float*)d_in[7];
    const float* g2  = (const float*)d_in[8];
    const float* be2 = (const float*)d_in[9];
    const float* W3  = (const float*)d_in[10];
    const float* b3  = (const float*)d_in[11];
    float* out = (float*)d_out;

    const int N = in_sizes[0] / IN_C;       // 200000
    const int E = in_sizes[1] / 2;          // 3200000
    const int* srcI = ei;
    const int* dstI = ei + E;

    // Workspace layout: dinv[N] | h[N*16] | agg[N*16] | stats[32]
    float* dinv  = (float*)d_ws;
    float* h     = dinv + N;
    float* agg   = h + (size_t)N * 16;
    float* stats = agg + (size_t)N * 16;

    const int NT = N * 16;
    const dim3 blk(256);
    const int gN   = (N + 255) / 256;
    const int gE   = (E + 255) / 256;
    const int gNT  = (NT + 255) / 256;
    const int gE4  = (E * 4 + 255) / 256;
    const int gTile = N / 16;               // 12500 exact
    const float invN = 1.0f / (float)N;

    // --- graph norms (once) ---
    init_deg_kernel<<<gN, blk, 0, stream>>>(dinv, N);
    count_deg_kernel<<<gE, blk, 0, stream>>>(dstI, dinv, E);
    deg_to_dinv_kernel<<<gN, blk, 0, stream>>>(dinv, N);

    // --- layer 1: GEMM(x,W1) -> agg -> BN+ReLU ---
    gemm16_wmma_kernel<IN_C><<<gTile, 32, 0, stream>>>(x, W1, h);
    init_agg_kernel<<<gNT, blk, 0, stream>>>(h, dinv, b1, agg, N);
    scatter_edges_kernel<<<gE4, blk, 0, stream>>>(srcI, dstI, h, dinv, agg, E);
    zero_stats_kernel<<<1, 32, 0, stream>>>(stats);
    bn_reduce_kernel<<<512, blk, 0, stream>>>(agg, stats, NT);
    bn_apply_relu_kernel<<<gNT, blk, 0, stream>>>(agg, stats, g1, be1, NT, invN);

    // --- layer 2 ---
    gemm16_wmma_kernel<HIDC><<<gTile, 32, 0, stream>>>(agg, W2, h);
    init_agg_kernel<<<gNT, blk, 0, stream>>>(h, dinv, b2, agg, N);
    scatter_edges_kernel<<<gE4, blk, 0, stream>>>(srcI, dstI, h, dinv, agg, E);
    zero_stats_kernel<<<1, 32, 0, stream>>>(stats);
    bn_reduce_kernel<<<512, blk, 0, stream>>>(agg, stats, NT);
    bn_apply_relu_kernel<<<gNT, blk, 0, stream>>>(agg, stats, g2, be2, NT, invN);

    // --- layer 3 (no BN/ReLU), writes d_out ---
    gemm16_wmma_kernel<HIDC><<<gTile, 32, 0, stream>>>(agg, W3, h);
    init_agg_kernel<<<gNT, blk, 0, stream>>>(h, dinv, b3, out, N);
    scatter_edges_kernel<<<gE4, blk, 0, stream>>>(srcI, dstI, h, dinv, out, E);
}